// BilateralFilter_51436528337261
// MI455X (gfx1250) — compile-verified
//
#include <hip/hip_runtime.h>
#include <cstdint>

// Bilateral filter, ksize=9, sigma=1.7, input [2,3,384,384] f32.
// out(p) = sum_t exp2(-((p_t-c)^2 + r_t^2)*C2) * p_t / sum_t (same weights)
// with C2 = 1/(2*sigma^2*ln2); all reference normalizations cancel in the ratio.
//
// Tap math (2 FMAs + 1 exp2): with S=sqrt(C2), ncs=-c*S:
//   e = fma(p, S, ncs)            // (p-c)*S
//   a = fma(-e, e, -r^2*C2)       // -((p-c)^2 + r^2)*C2
//   w = exp2(a)                   // v_exp_f32
// Taps processed in float2 pairs -> V_PK_FMA_F32 / V_PK_ADD_F32.

#define TILE   32
#define HALO   4
#define KS     9
#define LW     (TILE + 2 * HALO)      // 40
#define LDSN   (LW * LW)              // 1600 floats = 6.4 KB
#define NTHREADS 256

typedef float v2f __attribute__((ext_vector_type(2)));

// C2 = 1/(2*1.7^2 * ln2)
constexpr double C2d = 1.0 / (2.0 * 1.7 * 1.7 * 0.69314718055994530942);

// -((i-4)^2 + (j-4)^2) * C2, folded to a literal in the unrolled loops
__device__ __forceinline__ constexpr float nsd(int i, int j) {
    return (float)(-(double)((i - 4) * (i - 4) + (j - 4) * (j - 4)) * C2d);
}

// gfx1250 async copy: global memory -> LDS, per-lane 4B, tracked by ASYNCcnt.
__device__ __forceinline__ void async_g2lds_b32(uint32_t lds_addr, const float* gptr) {
    asm volatile("global_load_async_to_lds_b32 %0, %1, off"
                 :
                 : "v"(lds_addr), "v"((uint64_t)(uintptr_t)gptr)
                 : "memory");
}

__device__ __forceinline__ void wait_async_zero() {
    asm volatile("s_wait_asynccnt 0" ::: "memory");
}

__device__ __forceinline__ int reflect_idx(int p, int n) {
    // jnp.pad 'reflect' (no edge repeat); pad (=4) < n so one reflection suffices
    p = (p < 0) ? -p : p;
    p = (p >= n) ? (2 * n - 2 - p) : p;
    return p;
}

__global__ __launch_bounds__(NTHREADS)
void bilateral9_kernel(const float* __restrict__ x, float* __restrict__ out,
                       int H, int W) {
    __shared__ float tile[LDSN];

    const int plane = blockIdx.z;                 // 0..5 (B*C)
    const int bx = blockIdx.x * TILE;
    const int by = blockIdx.y * TILE;
    const float* __restrict__ src = x + (size_t)plane * H * W;

    const int tid = threadIdx.x;

    // ---- Stage 40x40 halo tile into LDS via CDNA5 async copies -------------
    // Generic pointer to __shared__ carries the LDS byte offset in low 32 bits.
    const uint32_t lds_base = (uint32_t)(uintptr_t)&tile[0];
    for (int idx = tid; idx < LDSN; idx += NTHREADS) {
        const int ly = idx / LW;
        const int lx = idx - ly * LW;
        const int gy = reflect_idx(by + ly - HALO, H);
        const int gx = reflect_idx(bx + lx - HALO, W);
        async_g2lds_b32(lds_base + (uint32_t)idx * 4u, src + (size_t)gy * W + gx);
    }
    wait_async_zero();     // this wave's async transfers landed in LDS
    __syncthreads();       // all waves' transfers visible

    // ---- Compute: 4 pixels per thread --------------------------------------
    const int tx  = tid & (TILE - 1);   // 0..31 -> stride-1 lanes, bank-conflict free
    const int tyb = tid >> 5;           // 0..7
    float* __restrict__ dst = out + (size_t)plane * H * W;

    const float S = __builtin_sqrtf((float)C2d);   // constant-folded
    const v2f  S2 = {S, S};

#pragma unroll 1
    for (int r = 0; r < 4; ++r) {
        const int row = tyb + r * 8;                       // tile-local output row
        const float c   = tile[(row + HALO) * LW + tx + HALO];
        const float ncs = -c * S;
        const v2f   ncs2 = {ncs, ncs};

        v2f   accw2  = {0.0f, 0.0f};
        v2f   accwp2 = {0.0f, 0.0f};
        float accw1  = 0.0f;
        float accwp1 = 0.0f;

#pragma unroll
        for (int i = 0; i < KS; ++i) {
            const float* __restrict__ rp = &tile[(row + i) * LW + tx];
            // 4 packed pairs (j=0..7)
#pragma unroll
            for (int j = 0; j < KS - 1; j += 2) {
                v2f p;
                p.x = rp[j];
                p.y = rp[j + 1];
                v2f e = p * S2 + ncs2;                       // v_pk_fma_f32
                const v2f k = {nsd(i, j), nsd(i, j + 1)};
                v2f a = k - e * e;                           // v_pk_fma (neg src)
                v2f w;
                w.x = __builtin_amdgcn_exp2f(a.x);           // v_exp_f32
                w.y = __builtin_amdgcn_exp2f(a.y);           // v_exp_f32
                accw2  = accw2 + w;                          // v_pk_add_f32
                accwp2 = accwp2 + w * p;                     // v_pk_fma_f32
            }
            // scalar tail (j=8)
            {
                const float p = rp[KS - 1];
                const float e = __builtin_fmaf(p, S, ncs);
                const float a = __builtin_fmaf(-e, e, nsd(i, KS - 1));
                const float w = __builtin_amdgcn_exp2f(a);
                accw1  += w;
                accwp1  = __builtin_fmaf(w, p, accwp1);
            }
        }
        const float accw  = (accw2.x + accw2.y) + accw1;
        const float accwp = (accwp2.x + accwp2.y) + accwp1;
        dst[(size_t)(by + row) * W + (bx + tx)] = accwp / accw;
    }
}

extern "C" void kernel_launch(void* const* d_in, const int* in_sizes, int n_in,
                              void* d_out, int out_size, void* d_ws, size_t ws_size,
                              hipStream_t stream) {
    (void)in_sizes; (void)n_in; (void)d_ws; (void)ws_size; (void)out_size;
    const float* x  = (const float*)d_in[0];   // [2,3,384,384] f32
    // d_in[1] is ksize (==9); hard-coded — cannot be read on host under graph capture.
    float* out = (float*)d_out;

    constexpr int H = 384, W = 384, PLANES = 2 * 3;
    dim3 grid(W / TILE, H / TILE, PLANES);     // 12 x 12 x 6
    dim3 block(NTHREADS, 1, 1);                // 8 wave32s
    bilateral9_kernel<<<grid, block, 0, stream>>>(x, out, H, W);
}